// TrunkFeatureLSTM_84567906058437
// MI455X (gfx1250) — compile-verified
//
#include <hip/hip_runtime.h>
#include <math.h>

// Problem constants
#define NB      1024           // batch
#define C_IN    22
#define HW      361            // 19*19
#define C_TRUNK 384
#define HID     128
#define NHIST   5
#define HIST_START 9
#define NGATE   (4*HID)        // 512
#define MSEQ    (NHIST*NB)     // 5120 rows of the input-transform GEMM

typedef __attribute__((ext_vector_type(2))) float v2f;
typedef __attribute__((ext_vector_type(8))) float v8f;

static __device__ __forceinline__ float sigmoidf_(float x) {
  return 1.0f / (1.0f + __expf(-x));
}

// WMMA helper: D = A(16x4) * B(4x16) + C, fp32.
static __device__ __forceinline__ v8f wmma_f32_k4(v2f a, v2f b, v8f c) {
  return __builtin_amdgcn_wmma_f32_16x16x4_f32(
      /*neg_a=*/false, a, /*neg_b=*/false, b,
      /*c_mod=*/(short)0, c, /*reuse_a=*/false, /*reuse_b=*/false);
}

// ---------------------------------------------------------------------------
// Kernel 1: per (n,t) -> sum & argmax over 361 spatial cells of history plane,
// then gather trunk features at argmax (masked), stored time-REVERSED.
// Fbuf layout: [s][n][c], s = NHIST-1-t.
// ---------------------------------------------------------------------------
__global__ __launch_bounds__(128) void prep_kernel(
    const float* __restrict__ input_spatial,
    const float* __restrict__ trunk_out,
    float* __restrict__ Fbuf) {
  const int n = blockIdx.x;
  const int t = blockIdx.y;
  const int tid = threadIdx.x;

  const float* plane = input_spatial + ((size_t)n * C_IN + (HIST_START + t)) * HW;

  __shared__ float s_max[128];
  __shared__ int   s_idx[128];
  __shared__ float s_sum[128];

  float lmax = -3.0e38f;
  int   lidx = 0;
  float lsum = 0.0f;
  for (int i = tid; i < HW; i += 128) {
    float v = plane[i];
    lsum += v;
    if (v > lmax) { lmax = v; lidx = i; }   // strictly-greater => first index kept
  }
  s_max[tid] = lmax; s_idx[tid] = lidx; s_sum[tid] = lsum;
  __syncthreads();

  for (int s = 64; s > 0; s >>= 1) {
    if (tid < s) {
      s_sum[tid] += s_sum[tid + s];
      float om = s_max[tid + s];
      int   oi = s_idx[tid + s];
      if (om > s_max[tid] || (om == s_max[tid] && oi < s_idx[tid])) {
        s_max[tid] = om; s_idx[tid] = oi;
      }
    }
    __syncthreads();
  }

  const int   idx  = s_idx[0];
  const float mask = (s_sum[0] > 0.5f) ? 1.0f : 0.0f;
  const int   srev = (NHIST - 1) - t;

  float* dst = Fbuf + ((size_t)srev * NB + n) * C_TRUNK;
  const float* src = trunk_out + (size_t)n * C_TRUNK * HW + idx;
  for (int c = tid; c < C_TRUNK; c += 128) {
    dst[c] = src[(size_t)c * HW] * mask;
  }
}

// ---------------------------------------------------------------------------
// Kernel 2: Gx = Fbuf(5120x384) @ w_ih^T(384x512) + (b_ih + b_hh)
// One wave per 16x16 output tile. fp32 WMMA, K-chunks of 4.
// ---------------------------------------------------------------------------
__global__ __launch_bounds__(32) void gemm_ih_kernel(
    const float* __restrict__ F,      // [5120][384]
    const float* __restrict__ w_ih,   // [512][384]
    const float* __restrict__ b_ih,
    const float* __restrict__ b_hh,
    float* __restrict__ Gx) {         // [5120][512]
  const int tile_m = blockIdx.x;      // 0..319
  const int tile_n = blockIdx.y;      // 0..31
  const int lane = threadIdx.x;
  const int half = lane >> 4;         // K sub-pair selector (0 -> K0,1 ; 1 -> K2,3)
  const int l15  = lane & 15;

  const float* Arow = F    + (size_t)(tile_m * 16 + l15) * C_TRUNK;
  const float* Brow = w_ih + (size_t)(tile_n * 16 + l15) * C_TRUNK; // B[k][n] = w_ih[n][k]

  v8f acc = {};
  for (int k = 0; k < C_TRUNK; k += 4) {
    const int k0 = k + half * 2;
    v2f a = *(const v2f*)(Arow + k0);
    v2f b = *(const v2f*)(Brow + k0);
    acc = wmma_f32_k4(a, b, acc);
  }

  const int ncol = tile_n * 16 + l15;
  const float bias = b_ih[ncol] + b_hh[ncol];
  const int mbase = tile_m * 16 + half * 8;
#pragma unroll
  for (int r = 0; r < 8; ++r) {
    Gx[(size_t)(mbase + r) * NGATE + ncol] = acc[r] + bias;
  }
}

// ---------------------------------------------------------------------------
// Kernel 3: zero-init of H0 and C (contiguous region)
// ---------------------------------------------------------------------------
__global__ __launch_bounds__(256) void zero_kernel(float* __restrict__ p, int nElem) {
  int i = blockIdx.x * 256 + threadIdx.x;
  if (i < nElem) p[i] = 0.0f;
}

// ---------------------------------------------------------------------------
// Kernel 4: one LSTM step.
//   gates = Hin(1024x128) @ w_hh^T(128x512) + Gx[s]
//   C = sigm(f)*C + sigm(i)*tanh(g);  Hout = sigm(o)*tanh(C)
// One wave owns a 16(batch) x 16(hidden) tile => computes i/f/g/o tiles
// (N offsets 0,128,256,384) with 4 WMMA accumulators, then does the cell
// update elementwise in-register (lane = N column, VGPR index = M row).
// ---------------------------------------------------------------------------
__global__ __launch_bounds__(32) void lstm_step_kernel(
    const float* __restrict__ Hin,    // [1024][128]
    float* __restrict__ Cst,          // [1024][128] (in-place, tile-exclusive)
    const float* __restrict__ w_hh,   // [512][128]
    const float* __restrict__ Gx,     // [1024][512] for this step
    float* __restrict__ Hout) {       // [1024][128]
  const int tile_m = blockIdx.x;      // 0..63
  const int tile_n = blockIdx.y;      // 0..7
  const int lane = threadIdx.x;
  const int half = lane >> 4;
  const int l15  = lane & 15;

  const int j = tile_n * 16 + l15;    // hidden unit column 0..127
  const float* Arow = Hin + (size_t)(tile_m * 16 + l15) * HID;
  const float* Bi = w_hh + (size_t)(0 * HID + j) * HID;
  const float* Bf = w_hh + (size_t)(1 * HID + j) * HID;
  const float* Bg = w_hh + (size_t)(2 * HID + j) * HID;
  const float* Bo = w_hh + (size_t)(3 * HID + j) * HID;

  v8f ai = {}, af = {}, ag = {}, ao = {};
  for (int k = 0; k < HID; k += 4) {
    const int k0 = k + half * 2;
    v2f a = *(const v2f*)(Arow + k0);
    ai = wmma_f32_k4(a, *(const v2f*)(Bi + k0), ai);
    af = wmma_f32_k4(a, *(const v2f*)(Bf + k0), af);
    ag = wmma_f32_k4(a, *(const v2f*)(Bg + k0), ag);
    ao = wmma_f32_k4(a, *(const v2f*)(Bo + k0), ao);
  }

  const int mbase = tile_m * 16 + half * 8;
#pragma unroll
  for (int r = 0; r < 8; ++r) {
    const size_t row = (size_t)(mbase + r);
    const float gi = ai[r] + Gx[row * NGATE + 0 * HID + j];
    const float gf = af[r] + Gx[row * NGATE + 1 * HID + j];
    const float gg = ag[r] + Gx[row * NGATE + 2 * HID + j];
    const float go = ao[r] + Gx[row * NGATE + 3 * HID + j];
    const float c_old = Cst[row * HID + j];
    const float c_new = sigmoidf_(gf) * c_old + sigmoidf_(gi) * tanhf(gg);
    const float h_new = sigmoidf_(go) * tanhf(c_new);
    Cst[row * HID + j]  = c_new;
    Hout[row * HID + j] = h_new;
  }
}

// ---------------------------------------------------------------------------
// Kernel 5: out = Hfinal(1024x128) @ w_proj^T(128x384) + b_proj
// ---------------------------------------------------------------------------
__global__ __launch_bounds__(32) void proj_kernel(
    const float* __restrict__ Hin,     // [1024][128]
    const float* __restrict__ w_proj,  // [384][128]
    const float* __restrict__ b_proj,  // [384]
    float* __restrict__ out) {         // [1024][384]
  const int tile_m = blockIdx.x;       // 0..63
  const int tile_n = blockIdx.y;       // 0..23
  const int lane = threadIdx.x;
  const int half = lane >> 4;
  const int l15  = lane & 15;

  const float* Arow = Hin    + (size_t)(tile_m * 16 + l15) * HID;
  const float* Brow = w_proj + (size_t)(tile_n * 16 + l15) * HID;

  v8f acc = {};
  for (int k = 0; k < HID; k += 4) {
    const int k0 = k + half * 2;
    v2f a = *(const v2f*)(Arow + k0);
    v2f b = *(const v2f*)(Brow + k0);
    acc = wmma_f32_k4(a, b, acc);
  }

  const int ncol = tile_n * 16 + l15;
  const float bias = b_proj[ncol];
  const int mbase = tile_m * 16 + half * 8;
#pragma unroll
  for (int r = 0; r < 8; ++r) {
    out[(size_t)(mbase + r) * C_TRUNK + ncol] = acc[r] + bias;
  }
}

// ---------------------------------------------------------------------------
extern "C" void kernel_launch(void* const* d_in, const int* in_sizes, int n_in,
                              void* d_out, int out_size, void* d_ws, size_t ws_size,
                              hipStream_t stream) {
  const float* input_spatial = (const float*)d_in[0];
  const float* trunk_out     = (const float*)d_in[1];
  const float* w_ih          = (const float*)d_in[2];
  const float* w_hh          = (const float*)d_in[3];
  const float* b_ih          = (const float*)d_in[4];
  const float* b_hh          = (const float*)d_in[5];
  const float* w_proj        = (const float*)d_in[6];
  const float* b_proj        = (const float*)d_in[7];
  float* out = (float*)d_out;

  // Workspace carve-up (floats):
  //   Fbuf : 5*1024*384  reversed gathered features
  //   Gx   : 5*1024*512  precomputed input-side gates (+biases)
  //   H0, C, H1 : 1024*128 each (H0/C zero-initialized contiguously)
  float* Fbuf = (float*)d_ws;
  float* Gx   = Fbuf + (size_t)NHIST * NB * C_TRUNK;
  float* H0   = Gx   + (size_t)NHIST * NB * NGATE;
  float* Cst  = H0   + (size_t)NB * HID;
  float* H1   = Cst  + (size_t)NB * HID;

  // 1) argmax/mask + gather (time-reversed)
  prep_kernel<<<dim3(NB, NHIST), 128, 0, stream>>>(input_spatial, trunk_out, Fbuf);

  // 2) big input-transform GEMM: M=5120, K=384, N=512 (fp32 WMMA)
  gemm_ih_kernel<<<dim3(MSEQ / 16, NGATE / 16), 32, 0, stream>>>(
      Fbuf, w_ih, b_ih, b_hh, Gx);

  // 3) zero H0 and C (contiguous)
  {
    const int nz = 2 * NB * HID;
    zero_kernel<<<(nz + 255) / 256, 256, 0, stream>>>(H0, nz);
  }

  // 4) 5 sequential LSTM steps (H ping-pong, C in place)
  float* hin = H0;
  float* hout = H1;
  for (int s = 0; s < NHIST; ++s) {
    lstm_step_kernel<<<dim3(NB / 16, HID / 16), 32, 0, stream>>>(
        hin, Cst, w_hh, Gx + (size_t)s * NB * NGATE, hout);
    float* tmp = hin; hin = hout; hout = tmp;
  }
  // final hidden state is in `hin` after the swap

  // 5) projection: M=1024, K=128, N=384
  proj_kernel<<<dim3(NB / 16, C_TRUNK / 16), 32, 0, stream>>>(
      hin, w_proj, b_proj, out);
}